// RuleLayer_66692252172661
// MI455X (gfx1250) — compile-verified
//
#include <hip/hip_runtime.h>

// out[b, i] = prod_{d=0..7} input[b, digit_d(i), d],  i in [0, 4^8), digit_0 = MSB.
// Split: hi[h] = prod d=0..3, lo[l] = prod d=4..7, out[b, h*256 + l] = hi[h]*lo[l].
// Per batch: 256x256 rank-1 outer product -> 16x16 tiles via V_WMMA_F32_16X16X4_F32
// with only the K=0 slice populated (A: K=0 column = hi segment, B: K=0 row = lo segment).

typedef __attribute__((ext_vector_type(2))) float v2f;
typedef __attribute__((ext_vector_type(8))) float v8f;

#define OUT_PER_BATCH 65536   // 4^8
#define IN_PER_BATCH  32      // M*D = 4*8

__global__ __launch_bounds__(256) void rule_outer_wmma_kernel(
    const float* __restrict__ in, float* __restrict__ out)
{
    __shared__ float s_in[IN_PER_BATCH];
    __shared__ float s_hi[256];
    __shared__ float s_lo[256];

    const int b = blockIdx.x;
    const int t = threadIdx.x;            // 0..255

    // Stage the 32 input floats for this batch.
    if (t < IN_PER_BATCH) s_in[t] = in[b * IN_PER_BATCH + t];
    __syncthreads();

    // Thread t owns base-4 digits (m0,m1,m2,m3) of index t (m0 = MSB).
    const int m0 = (t >> 6) & 3;
    const int m1 = (t >> 4) & 3;
    const int m2 = (t >> 2) & 3;
    const int m3 =  t       & 3;
    // input layout: s_in[m*8 + d]
    s_hi[t] = s_in[m0 * 8 + 0] * s_in[m1 * 8 + 1] * s_in[m2 * 8 + 2] * s_in[m3 * 8 + 3];
    s_lo[t] = s_in[m0 * 8 + 4] * s_in[m1 * 8 + 5] * s_in[m2 * 8 + 6] * s_in[m3 * 8 + 7];
    __syncthreads();

    const int wave   = t >> 5;            // 0..7
    const int lane   = t & 31;
    const int lane15 = lane & 15;
    const bool lower = (lane < 16);       // K=0 slots of A/B live in lanes 0..15 of VGPR0

    float* outb = out + (size_t)b * OUT_PER_BATCH;

    // Each wave: 2 tile-rows (h) x 16 tile-cols (l) = 32 WMMA tiles of 16x16.
    for (int hr = 0; hr < 2; ++hr) {
        const int h0 = (wave * 2 + hr) * 16;

        // A-matrix (16x4 f32): K=0 column = hi[h0..h0+15]; all other K slots zero.
        // ISA layout: A VGPR0 lanes0-15 = K0 (M=lane), lanes16-31 = K2; VGPR1 = K1/K3.
        const float hv = s_hi[h0 + lane15];
        v2f a;
        a.x = lower ? hv : 0.0f;          // select, not branch: EXEC stays all-ones
        a.y = 0.0f;

        for (int lt = 0; lt < 16; ++lt) {
            const int l0 = lt * 16;

            // B-matrix (4x16 f32): K=0 row = lo[l0..l0+15]; other K slots zero.
            const float lv = s_lo[l0 + lane15];
            v2f bm;
            bm.x = lower ? lv : 0.0f;
            bm.y = 0.0f;

            v8f c = {};
            // D = A x B + 0  (8 args: neg_a, A, neg_b, B, c_mod, C, reuse_a, reuse_b)
            c = __builtin_amdgcn_wmma_f32_16x16x4_f32(
                    false, a, false, bm, (short)0, c, false, false);

            // C/D layout: VGPR r -> row (h0 + r) for lanes 0-15, row (h0 + 8 + r) for
            // lanes 16-31; col = l0 + (lane & 15). Half-wave rows are 64B contiguous.
            float* p = outb + (size_t)(h0 + (lower ? 0 : 8)) * 256 + (l0 + lane15);
            #pragma unroll
            for (int r = 0; r < 8; ++r) {
                p[r * 256] = c[r];        // 8x global_store_b32, immediate offsets
            }
        }
    }
}

extern "C" void kernel_launch(void* const* d_in, const int* in_sizes, int n_in,
                              void* d_out, int out_size, void* d_ws, size_t ws_size,
                              hipStream_t stream) {
    const float* in = (const float*)d_in[0];
    float* out = (float*)d_out;
    const int B = in_sizes[0] / IN_PER_BATCH;   // 512
    dim3 grid(B), block(256);
    rule_outer_wmma_kernel<<<grid, block, 0, stream>>>(in, out);
}